// RobustVectorQuantizer_4724464025935
// MI455X (gfx1250) — compile-verified
//
#include <hip/hip_runtime.h>
#include <math.h>

// ---------------------------------------------------------------------------
// VQ nearest-codeword (256 codewords, dim 2) for MI455X / gfx1250.
//
// score(M,N) = |c_N|^2 - 2 * z_M . c_N   computed by V_WMMA_F32_16X16X4_F32:
//   A (16x4, M=z)  row = (zx, zy, 1, 0)
//   B (4x16, N=cw) col = (-2cx, -2cy, |c|^2, 0)
// 16 B-tiles (= all 256 codewords) are loop-invariant VGPR residents.
// Argmin: pack 8-bit codeword id into score mantissa LSBs, fmin/min3 tree.
// z loads are software-pipelined one group ahead so the WMMA+select block of
// group i hides the load latency of group i+1.
// ---------------------------------------------------------------------------

typedef float v2f __attribute__((ext_vector_type(2)));
typedef float v8f __attribute__((ext_vector_type(8)));

#define NZ       (1024 * 1024)      // number of 2-d z vectors
#define NELEM    (2 * NZ)           // total float elements of z / z_q
#define NUM_EMB  256
#define NGROUPS  (NZ / 16)          // 65536 groups of 16 z-vectors
#define TPB      256
#define WPB      (TPB / 32)         // 8 waves per block (wave32)
#define BLOCKS   1024
#define NWAVES   (BLOCKS * WPB)     // 8192
#define GPW      (NGROUPS / NWAVES) // 8 groups per wave, exact

__global__ __launch_bounds__(TPB) void vq_main(const float* __restrict__ z,
                                               const float* __restrict__ cb,
                                               float* __restrict__ out,
                                               float* __restrict__ partial)
{
    __shared__ float2 sCB[NUM_EMB];

    const int tid = threadIdx.x;
    // Stage codebook to LDS (2 KB): one float2 per thread.
    sCB[tid] = ((const float2*)cb)[tid];
    __syncthreads();

    const int  lane = tid & 31;
    const int  wid  = blockIdx.x * WPB + (tid >> 5);
    const bool lo   = lane < 16;       // lanes 0-15 carry K={0,1}; 16-31 carry K={2,3}
    const int  col  = lane & 15;       // N / M position within a 16-wide tile

    // Build 16 loop-invariant B tiles + packed index vectors (VGPR resident).
    float    B0[16], B1[16];
    unsigned IV[16];
#pragma unroll
    for (int j = 0; j < 16; ++j) {
        float2 c = sCB[j * 16 + col];
        B0[j] = lo ? (-2.0f * c.x) : (c.x * c.x + c.y * c.y); // K0 row / K2 row
        B1[j] = lo ? (-2.0f * c.y) : 0.0f;                    // K1 row / K3 row
        IV[j] = (unsigned)(j * 16 + col);                     // codeword id per lane
    }

    const float2* z2   = (const float2*)z;
    float2*       out2 = (float2*)out;
    float         acc  = 0.0f;

    const int g0 = wid * GPW;
    // Pipeline prologue: load group 0's z tile (lanes 16-31 read a duplicate
    // in-bounds address; constants are selected in afterwards).
    float2 zv = z2[g0 * 16 + col];

#pragma unroll 1
    for (int i = 0; i < GPW; ++i) {
        const int g = g0 + i;

        // Issue next group's load now; its wait lands after this group's
        // WMMA + select block (~300 instructions of cover).
        const int gn = (i + 1 < GPW) ? (g + 1) : g;   // clamped, always in bounds
        float2 zv_next = z2[gn * 16 + col];

        v2f a;
        a.x = lo ? zv.x : 1.0f;   // A VGPR0: K=0 (zx) | K=2 (1.0)
        a.y = lo ? zv.y : 0.0f;   // A VGPR1: K=1 (zy) | K=3 (0.0)

        // 16 WMMAs -> running packed min over all 256 codewords.
        v8f m = {};
#pragma unroll
        for (int j = 0; j < 16; ++j) {
            v2f b;
            b.x = B0[j];
            b.y = B1[j];
            v8f czero = {};
            v8f d = __builtin_amdgcn_wmma_f32_16x16x4_f32(
                false, a, false, b, (short)0, czero, false, false);
#pragma unroll
            for (int v = 0; v < 8; ++v) {
                unsigned p  = (__float_as_uint(d[v]) & 0xFFFFFF00u) | IV[j];
                float    pf = __uint_as_float(p);
                m[v] = (j == 0) ? pf : fminf(m[v], pf);
            }
        }

        // Min across the 16 N-lanes of each half (rows stay per-VGPR).
#pragma unroll
        for (int v = 0; v < 8; ++v) {
            float t = m[v];
            t = fminf(t, __shfl_xor(t, 1, 32));
            t = fminf(t, __shfl_xor(t, 2, 32));
            t = fminf(t, __shfl_xor(t, 4, 32));
            t = fminf(t, __shfl_xor(t, 8, 32));
            m[v] = t;
        }

        // Redistribute: lane l (0..15) gets the winner of row l.
        // Row r lives in VGPR (r&7) of half (r>>3). cndmask tree + bpermute.
        const int t3 = lane & 7;
        float s0 = (t3 & 1) ? m[1] : m[0];
        float s1 = (t3 & 1) ? m[3] : m[2];
        float s2 = (t3 & 1) ? m[5] : m[4];
        float s3 = (t3 & 1) ? m[7] : m[6];
        float u0 = (t3 & 2) ? s1 : s0;
        float u1 = (t3 & 2) ? s3 : s2;
        float w  = (t3 & 4) ? u1 : u0;
        const int srcLane = (lane & 7) | ((lane & 8) << 1);
        float win = __shfl(w, srcLane, 32);

        const unsigned idx = __float_as_uint(win) & 0xFFu;
        float2 c = sCB[idx];

        if (lo) {
            out2[g * 16 + lane] = c;                 // z_q, coalesced b64 stores
            float dx = zv.x - c.x;
            float dy = zv.y - c.y;
            acc += dx * dx + dy * dy;                // exact per-z distance
        }

        zv = zv_next;                                // rotate pipeline
    }

    // Deterministic wave reduction; fixed partial slot per wave.
    acc += __shfl_xor(acc, 1, 32);
    acc += __shfl_xor(acc, 2, 32);
    acc += __shfl_xor(acc, 4, 32);
    acc += __shfl_xor(acc, 8, 32);
    acc += __shfl_xor(acc, 16, 32);
    if (lane == 0) partial[wid] = acc;
}

__global__ __launch_bounds__(TPB) void vq_finalize(const float* __restrict__ partial,
                                                   float* __restrict__ out)
{
    __shared__ float s[TPB];
    const int t = threadIdx.x;
    float a = 0.0f;
#pragma unroll
    for (int k = 0; k < NWAVES / TPB; ++k)           // fixed order per thread
        a += partial[t + k * TPB];
    s[t] = a;
    __syncthreads();
    for (int off = TPB / 2; off > 0; off >>= 1) {    // fixed-order tree
        if (t < off) s[t] += s[t + off];
        __syncthreads();
    }
    if (t == 0)
        out[NELEM] = 1.25f * s[0] / (float)NELEM;    // (1 + BETA) * mean((zq-z)^2)
}

extern "C" void kernel_launch(void* const* d_in, const int* in_sizes, int n_in,
                              void* d_out, int out_size, void* d_ws, size_t ws_size,
                              hipStream_t stream)
{
    (void)in_sizes; (void)n_in; (void)out_size; (void)ws_size;
    const float* z  = (const float*)d_in[0];   // [1024,1024,2] f32
    const float* cb = (const float*)d_in[1];   // [256,2] f32
    float* out      = (float*)d_out;           // z_q (2097152 f32) + loss (1 f32)
    float* partial  = (float*)d_ws;            // 8192 f32 wave partials (32 KB)

    vq_main<<<BLOCKS, TPB, 0, stream>>>(z, cb, out, partial);
    vq_finalize<<<1, TPB, 0, stream>>>(partial, out);
}